// pendelum3dof_16707422781835
// MI455X (gfx1250) — compile-verified
//
#include <hip/hip_runtime.h>
#include <stdint.h>

// ---------------------------------------------------------------------------
// Triple-pendulum Hamiltonian dynamics, closed form.
// Memory-bound: ~100MB traffic -> ~4.3us floor @ 23.3TB/s.
// Strategy: 256 rows/block staged through LDS with CDNA5 async global<->LDS
// copies (ASYNCcnt) for fully-coalesced b64 global traffic; bank-conflict-free
// ds reads (stride 3 float2 => 6 dwords/row, 3 invertible mod 32).
// ---------------------------------------------------------------------------

#define GRAV 9.81f
#define ROWS_PER_BLK 256

__device__ __forceinline__ uint32_t lds_addr_u32(const void* p) {
  // Low 32 bits of a generic pointer to a __shared__ object are the LDS byte
  // offset on gfx1250 (aperture bits live in the high half).
  return (uint32_t)(uintptr_t)p;
}

__device__ __forceinline__ void wait_asynccnt0() {
#if __has_builtin(__builtin_amdgcn_s_wait_asynccnt)
  __builtin_amdgcn_s_wait_asynccnt(0);
#else
  asm volatile("s_wait_asynccnt 0" ::: "memory");
#endif
}

// Async copy 8 bytes: global -> LDS (tracked by ASYNCcnt).
__device__ __forceinline__ void async_g2l_b64(const void* gptr, uint32_t lds) {
  asm volatile("global_load_async_to_lds_b64 %0, %1, off"
               :
               : "v"(lds), "v"(gptr)
               : "memory");
}

// Async copy 8 bytes: LDS -> global (tracked by ASYNCcnt).
__device__ __forceinline__ void async_l2g_b64(void* gptr, uint32_t lds) {
  asm volatile("global_store_async_from_lds_b64 %0, %1, off"
               :
               : "v"(gptr), "v"(lds)
               : "memory");
}

// Closed-form d/dt [theta; p] for the 3-link pendulum (M1..3=L1..3=1, G=9.81).
__device__ __forceinline__ void pend3_compute(const float in6[6], float o6[6]) {
  const float a = in6[0], b = in6[1], c = in6[2];
  const float p0 = in6[3], p1 = in6[4], p2 = in6[5];

  float sa, ca, sb, cb, sc, cc;
  __sincosf(a, &sa, &ca);
  __sincosf(b, &sb, &cb);
  __sincosf(c, &sc, &cc);

  const float cab = ca * cb + sa * sb, sab = sa * cb - ca * sb; // cos/sin(a-b)
  const float cac = ca * cc + sa * sc, sac = sa * cc - ca * sc; // cos/sin(a-c)
  const float cbc = cb * cc + sb * sc, sbc = sb * cc - cb * sc; // cos/sin(b-c)

  // M = [[3, d, e],[d, 2, f],[e, f, 1]]
  const float d = 2.0f * cab, e = cac, f = cbc;

  // Adjugate (symmetric) and determinant.
  const float A00 = 2.0f - f * f;
  const float A01 = e * f - d;
  const float A02 = d * f - 2.0f * e;
  const float A11 = 3.0f - e * e;
  const float A12 = d * e - 3.0f * f;
  const float A22 = 6.0f - d * d;
  const float det = 3.0f * A00 + d * A01 + e * A02; // SPD -> well-conditioned
  const float inv = 1.0f / det;

  // theta_dot = M^-1 p
  const float u = (A00 * p0 + A01 * p1 + A02 * p2) * inv;
  const float v = (A01 * p0 + A11 * p1 + A12 * p2) * inv;
  const float w = (A02 * p0 + A12 * p1 + A22 * p2) * inv;

  const float tuv = u * v * sab;
  const float tuw = u * w * sac;
  const float tvw = v * w * sbc;

  o6[0] = u;
  o6[1] = v;
  o6[2] = w;
  // pdot = -dH/dtheta
  o6[3] = -(2.0f * tuv + tuw + 3.0f * GRAV * sa);
  o6[4] = 2.0f * tuv - tvw - 2.0f * GRAV * sb;
  o6[5] = tuw + tvw - GRAV * sc;
}

__global__ __launch_bounds__(ROWS_PER_BLK) void pend3_block_kernel(
    const float* __restrict__ y, float* __restrict__ out) {
  __shared__ __align__(16) float s_in[ROWS_PER_BLK * 6];
  __shared__ __align__(16) float s_out[ROWS_PER_BLK * 6];

  const int tid = threadIdx.x;
  const size_t base = (size_t)blockIdx.x * (ROWS_PER_BLK * 6);
  const float* gin = y + base;
  float* gout = out + base;

  // ---- Stage 6144B of input into LDS: 768 coalesced async b64 chunks ----
#pragma unroll
  for (int i = 0; i < 3; ++i) {
    const int idx = tid + i * ROWS_PER_BLK; // float2 chunk id, 0..767
    async_g2l_b64(gin + idx * 2, lds_addr_u32(&s_in[idx * 2]));
  }
  wait_asynccnt0();
  __syncthreads();

  // ---- Per-row compute; stride-3 float2 LDS reads are bank-conflict-free ----
  float in6[6], o6[6];
  {
    const float2* r2 = reinterpret_cast<const float2*>(s_in) + tid * 3;
    const float2 q0 = r2[0], q1 = r2[1], q2 = r2[2];
    in6[0] = q0.x; in6[1] = q0.y; in6[2] = q1.x;
    in6[3] = q1.y; in6[4] = q2.x; in6[5] = q2.y;
  }
  pend3_compute(in6, o6);
  {
    float2* w2 = reinterpret_cast<float2*>(s_out) + tid * 3;
    w2[0] = make_float2(o6[0], o6[1]);
    w2[1] = make_float2(o6[2], o6[3]);
    w2[2] = make_float2(o6[4], o6[5]);
  }
  __syncthreads();

  // ---- Stream results back: coalesced async b64 chunks ----
#pragma unroll
  for (int i = 0; i < 3; ++i) {
    const int idx = tid + i * ROWS_PER_BLK;
    async_l2g_b64(gout + idx * 2, lds_addr_u32(&s_out[idx * 2]));
  }
  wait_asynccnt0(); // S_ENDPGM also implicitly waits idle
}

// Tail path for row counts not divisible by ROWS_PER_BLK (not hit for 2M rows,
// but keeps kernel_launch correct for any size).
__global__ void pend3_tail_kernel(const float* __restrict__ y,
                                  float* __restrict__ out, long long row0,
                                  long long nrows) {
  const long long r =
      row0 + (long long)blockIdx.x * blockDim.x + threadIdx.x;
  if (r >= nrows) return;
  const float* p = y + r * 6;
  float in6[6], o6[6];
#pragma unroll
  for (int k = 0; k < 6; ++k) in6[k] = p[k];
  pend3_compute(in6, o6);
  float* q = out + r * 6;
#pragma unroll
  for (int k = 0; k < 6; ++k) q[k] = o6[k];
}

extern "C" void kernel_launch(void* const* d_in, const int* in_sizes, int n_in,
                              void* d_out, int out_size, void* d_ws,
                              size_t ws_size, hipStream_t stream) {
  // d_in[0] = t (unused scalar), d_in[1] = y, float32, (nrows, 6) row-major.
  const float* y = (const float*)d_in[1];
  float* out = (float*)d_out;

  const long long nrows = (long long)in_sizes[1] / 6;
  const long long full_blocks = nrows / ROWS_PER_BLK;
  const long long tail_rows = nrows - full_blocks * ROWS_PER_BLK;

  if (full_blocks > 0) {
    pend3_block_kernel<<<(uint32_t)full_blocks, ROWS_PER_BLK, 0, stream>>>(y,
                                                                           out);
  }
  if (tail_rows > 0) {
    const uint32_t tb =
        (uint32_t)((tail_rows + ROWS_PER_BLK - 1) / ROWS_PER_BLK);
    pend3_tail_kernel<<<tb, ROWS_PER_BLK, 0, stream>>>(
        y, out, full_blocks * ROWS_PER_BLK, nrows);
  }
}